// DeformableAttention1D_63556926046317
// MI455X (gfx1250) — compile-verified
//
#include <hip/hip_runtime.h>
#include <math.h>
#include <stdint.h>

typedef __attribute__((ext_vector_type(2))) float v2f;
typedef __attribute__((ext_vector_type(8))) float v8f;

#define BB    4
#define NN    4096
#define DD    256
#define HH    8
#define DHH   64
#define PP    4
#define INNERD 512
#define LMM   8192

// ---------------------------------------------------------------------------
// Kernel 0: time-embed add + two layernorms.
// ---------------------------------------------------------------------------
__global__ void k_prep(const float* __restrict__ x, const float* __restrict__ prev_x,
                       const float* __restrict__ te,
                       const float* __restrict__ qs, const float* __restrict__ qb,
                       const float* __restrict__ ms, const float* __restrict__ mb,
                       float* __restrict__ qin, float* __restrict__ mem)
{
    const int rid = blockIdx.x;           // b*N + n
    const int b   = rid / NN;
    const int n   = rid % NN;
    const int t   = threadIdx.x;          // 0..255 == d

    const float vp = prev_x[(size_t)rid * DD + t] + te[t];
    const float vx = x[(size_t)rid * DD + t] + te[DD + t];

    __shared__ float red[256];
    float sums[4];
    const float in4[4] = { vp, vp * vp, vx, vx * vx };
    #pragma unroll
    for (int q = 0; q < 4; ++q) {
        red[t] = in4[q];
        __syncthreads();
        for (int s = 128; s > 0; s >>= 1) {
            if (t < s) red[t] += red[t + s];
            __syncthreads();
        }
        sums[q] = red[0];
        __syncthreads();
    }

    const float mq  = (sums[0] + sums[2]) * (1.0f / 512.0f);
    const float vq  = (sums[1] + sums[3]) * (1.0f / 512.0f) - mq * mq;
    const float rsq = rsqrtf(vq + 1e-5f);
    qin[(size_t)rid * INNERD + t]      = (vp - mq) * rsq * qs[t]      + qb[t];
    qin[(size_t)rid * INNERD + DD + t] = (vx - mq) * rsq * qs[DD + t] + qb[DD + t];

    const float mp  = sums[0] * (1.0f / 256.0f);
    const float vpv = sums[1] * (1.0f / 256.0f) - mp * mp;
    const float rsp = rsqrtf(vpv + 1e-5f);
    mem[((size_t)b * LMM + n) * DD + t] = (vp - mp) * rsp * ms[t] + mb[t];

    const float mx  = sums[2] * (1.0f / 256.0f);
    const float vxv = sums[3] * (1.0f / 256.0f) - mx * mx;
    const float rsx = rsqrtf(vxv + 1e-5f);
    mem[((size_t)b * LMM + NN + n) * DD + t] = (vx - mx) * rsx * ms[t] + mb[t];
}

// ---------------------------------------------------------------------------
// f32 WMMA GEMM with async-LDS double-buffered B panel:
//   C[M x Nc] = A[M x K] @ W[K x Nc] + bias (+ resid + te1)
//
// Block = 256 threads (8 waves); block tile 256(M) x 64(N); K chunk = 32.
// B panel (32 x 64 f32, 8 KB) staged into LDS with per-lane
// global_load_async_to_lds_b32 (ASYNCcnt), double-buffered so chunk i+1
// streams in while chunk i feeds the WMMAs.
// LDS layout pairs K-rows (k even/odd interleaved) so each 16x4 B fragment
// is one ds_load_b64 per lane.
// Each wave: 32(M) x 64(N) tile = 2 A fragments, 4 B fragments,
// 8 x V_WMMA_F32_16X16X4_F32 per K-step of 4.
// ---------------------------------------------------------------------------
__global__ void k_gemm_wmma(const float* __restrict__ A, const float* __restrict__ W,
                            const float* __restrict__ bias, float* __restrict__ C,
                            int M, int K, int Nc,
                            const float* __restrict__ resid,
                            const float* __restrict__ te1)
{
    __shared__ float tileB[2][32 * 64];   // paired layout: [kk>>1][col][kk&1]

    const int tid   = threadIdx.x;        // 0..255
    const int wave  = tid >> 5;           // 0..7
    const int lane  = tid & 31;
    const int row   = lane & 15;
    const int khalf = lane >> 4;          // 0 or 1
    const int m0    = blockIdx.y * 256 + wave * 32;
    const int n0    = blockIdx.x * 64;

    const float* __restrict__ Arow0 = A + (size_t)(m0 + row) * K;
    const float* __restrict__ Arow1 = A + (size_t)(m0 + 16 + row) * K;

    // Stage one 32x64 B chunk (rows kc..kc+31, cols n0..n0+63) into LDS buf.
    auto stage = [&](int kc, float* buf) {
        #pragma unroll
        for (int j = 0; j < 8; ++j) {
            const int f   = j * 256 + tid;       // 0..2047 flat float index
            const int kk  = f >> 6;              // 0..31
            const int col = f & 63;              // 0..63
            const float* g = W + (size_t)(kc + kk) * Nc + n0 + col;
            const uint32_t l =
                (uint32_t)(uintptr_t)(buf + ((kk >> 1) * 128 + col * 2 + (kk & 1)));
            asm volatile("global_load_async_to_lds_b32 %0, %1, off"
                         :: "v"(l), "v"(g) : "memory");
        }
    };

    v8f c[2][4] = {};
    const int nch = K >> 5;               // K / 32

    stage(0, tileB[0]);                   // prologue

    for (int ic = 0; ic < nch; ++ic) {
        // Drain this wave's outstanding asyncs (== chunk ic), then barrier so
        // every wave's share of the panel is visible (and every wave is done
        // reading the buffer we are about to overwrite).
        asm volatile("s_wait_asynccnt 0" ::: "memory");
        __syncthreads();

        if (ic + 1 < nch) stage((ic + 1) << 5, tileB[(ic + 1) & 1]);

        const float* __restrict__ bufc = tileB[ic & 1];
        const int kc = ic << 5;

        #pragma unroll
        for (int k8 = 0; k8 < 8; ++k8) {
            const int k = kc + k8 * 4;
            v2f a0, a1;
            a0.x = Arow0[k + 2 * khalf + 0];
            a0.y = Arow0[k + 2 * khalf + 1];
            a1.x = Arow1[k + 2 * khalf + 0];
            a1.y = Arow1[k + 2 * khalf + 1];
            const int pr = k8 * 2 + khalf;       // paired-row index
            #pragma unroll
            for (int tn = 0; tn < 4; ++tn) {
                const v2f bf =
                    *(const v2f*)&bufc[pr * 128 + (tn * 16 + row) * 2];
                c[0][tn] = __builtin_amdgcn_wmma_f32_16x16x4_f32(
                               false, a0, false, bf, (short)0, c[0][tn], false, false);
                c[1][tn] = __builtin_amdgcn_wmma_f32_16x16x4_f32(
                               false, a1, false, bf, (short)0, c[1][tn], false, false);
            }
        }
    }

    #pragma unroll
    for (int fg = 0; fg < 2; ++fg) {
        #pragma unroll
        for (int tn = 0; tn < 4; ++tn) {
            const int col  = n0 + tn * 16 + row;
            const float bcol = bias[col];
            #pragma unroll
            for (int r = 0; r < 8; ++r) {
                const int mrow = m0 + fg * 16 + r + 8 * khalf;
                float val = c[fg][tn][r] + bcol;
                if (resid != nullptr) {
                    val += resid[(size_t)mrow * Nc + col] + te1[col];
                }
                C[(size_t)mrow * Nc + col] = val;
            }
        }
    }
}

// ---------------------------------------------------------------------------
// Kernel 3: per-row head projections + position math + bilinear gather.
// ---------------------------------------------------------------------------
__global__ void k_sample(const float* __restrict__ q, const float* __restrict__ v,
                         const float* __restrict__ Wr, const float* __restrict__ br,
                         const float* __restrict__ Wd, const float* __restrict__ bd,
                         const float* __restrict__ Ww, const float* __restrict__ bw,
                         const float* __restrict__ base_off,
                         const float* __restrict__ max_off,
                         float* __restrict__ y, float* __restrict__ offs_out)
{
    const int rid = blockIdx.x;          // b*N + n
    const int b   = rid / NN;
    const int t   = threadIdx.x;         // 0..127

    __shared__ float qrow[INNERD];
    __shared__ float raw[72];            // [0:8)=r, [8:40)=delta, [40:72)=w
    __shared__ float attnS[HH * PP];
    __shared__ int   leftS[HH * PP];
    __shared__ int   rightS[HH * PP];
    __shared__ float fracS[HH * PP];

    #pragma unroll
    for (int i = 0; i < 4; ++i)
        qrow[t + 128 * i] = q[(size_t)rid * INNERD + t + 128 * i];
    __syncthreads();

    if (t < 72) {
        float acc;
        if (t < 8) {
            acc = br[t];
            for (int k = 0; k < INNERD; ++k) acc += qrow[k] * Wr[k * HH + t];
        } else if (t < 40) {
            const int j = t - 8;
            acc = bd[j];
            for (int k = 0; k < INNERD; ++k) acc += qrow[k] * Wd[k * (HH * PP) + j];
        } else {
            const int j = t - 40;
            acc = bw[j];
            for (int k = 0; k < INNERD; ++k) acc += qrow[k] * Ww[k * (HH * PP) + j];
        }
        raw[t] = acc;
    }
    __syncthreads();

    if (t < HH) {
        const int h = t;
        const float refv = 1.0f / (1.0f + __expf(-raw[h]));
        const float mo   = max_off[0];

        float aw[PP], pos[PP], dwin[PP];
        float wmax = -1e30f;
        #pragma unroll
        for (int p = 0; p < PP; ++p) wmax = fmaxf(wmax, raw[40 + h * PP + p]);
        float wsum = 0.0f;
        #pragma unroll
        for (int p = 0; p < PP; ++p) {
            aw[p] = __expf(raw[40 + h * PP + p] - wmax);
            wsum += aw[p];
        }
        float msum = 0.0f;
        #pragma unroll
        for (int p = 0; p < PP; ++p) {
            const float dl = raw[8 + h * PP + p];
            dwin[p] = base_off[h * PP + p] + tanhf(dl) * mo;
            float pp = refv * (float)(LMM - 1) + dwin[p];
            if (isnan(pp)) pp = 0.0f;
            pos[p] = pp;
            aw[p] /= wsum;
            const bool valid = (pp >= 0.0f) && (pp <= (float)(LMM - 1));
            aw[p] = valid ? aw[p] : 0.0f;
            msum += aw[p];
            offs_out[(size_t)rid * (HH * PP) + h * PP + p] = dwin[p] / (float)(LMM - 1);
        }
        const float inv = 1.0f / (msum + 1e-6f);
        #pragma unroll
        for (int p = 0; p < PP; ++p) {
            const float pc = fminf(fmaxf(pos[p], 0.0f), (float)(LMM - 1));
            const int   l  = (int)floorf(pc);
            const int   r  = min(l + 1, LMM - 1);
            attnS[h * PP + p] = aw[p] * inv;
            leftS[h * PP + p] = l;
            rightS[h * PP + p] = r;
            fracS[h * PP + p] = pc - (float)l;
        }
    }
    __syncthreads();

    const size_t vb = (size_t)b * LMM * INNERD;
    for (int o = t; o < INNERD; o += 128) {
        const int h = o >> 6;
        const int d = o & 63;
        float acc = 0.0f;
        #pragma unroll
        for (int p = 0; p < PP; ++p) {
            const int   idx = h * PP + p;
            const float f   = fracS[idx];
            const float vl  = v[vb + (size_t)leftS[idx]  * INNERD + h * DHH + d];
            const float vr  = v[vb + (size_t)rightS[idx] * INNERD + h * DHH + d];
            acc += attnS[idx] * ((1.0f - f) * vl + f * vr);
        }
        y[(size_t)rid * INNERD + o] = acc;
    }
}

// ---------------------------------------------------------------------------
extern "C" void kernel_launch(void* const* d_in, const int* in_sizes, int n_in,
                              void* d_out, int out_size, void* d_ws, size_t ws_size,
                              hipStream_t stream)
{
    const float* x      = (const float*)d_in[0];
    const float* prev_x = (const float*)d_in[1];
    const float* te     = (const float*)d_in[2];
    const float* qs     = (const float*)d_in[3];
    const float* qb     = (const float*)d_in[4];
    const float* ms     = (const float*)d_in[5];
    const float* mb     = (const float*)d_in[6];
    const float* Wq     = (const float*)d_in[7];
    const float* bq     = (const float*)d_in[8];
    const float* Wv     = (const float*)d_in[9];
    const float* bv     = (const float*)d_in[10];
    const float* Wr     = (const float*)d_in[11];
    const float* br     = (const float*)d_in[12];
    const float* Wd     = (const float*)d_in[13];
    const float* bd     = (const float*)d_in[14];
    const float* Ww     = (const float*)d_in[15];
    const float* bw     = (const float*)d_in[16];
    const float* Wo     = (const float*)d_in[17];
    const float* bo     = (const float*)d_in[18];
    const float* boff   = (const float*)d_in[19];
    const float* moff   = (const float*)d_in[20];

    float* out_main = (float*)d_out;                              // (B,N,D)
    float* out_offs = (float*)d_out + (size_t)BB * NN * DD;       // (B,N,H,P)

    // Workspace layout (128 MB total), buffers reused stream-ordered:
    //   [0,   32MB): qin (B,N,512)      -> later reused as y (B,N,512)
    //   [32,  64MB): mem (B,Lm,256)     -> later reused as q (B,N,512)
    //   [64, 128MB): v   (B,Lm,512)
    char* ws    = (char*)d_ws;
    float* qin  = (float*)(ws);
    float* mem  = (float*)(ws + (size_t)32 * 1024 * 1024);
    float* vmt  = (float*)(ws + (size_t)64 * 1024 * 1024);
    float* qbuf = mem;   // reuse after mem is consumed
    float* ybuf = qin;   // reuse after qin is consumed

    const int ROWS    = BB * NN;     // 16384
    const int MEMROWS = BB * LMM;    // 32768

    // 1) time-embed + layernorms
    k_prep<<<ROWS, 256, 0, stream>>>(x, prev_x, te, qs, qb, ms, mb, qin, mem);

    // 2) v = mem @ Wv + bv     (32768 x 512, K=256)
    k_gemm_wmma<<<dim3(INNERD / 64, MEMROWS / 256), 256, 0, stream>>>(
        mem, Wv, bv, vmt, MEMROWS, DD, INNERD, nullptr, nullptr);

    // 3) q = qin @ Wq + bq     (16384 x 512, K=512)   [writes over mem]
    k_gemm_wmma<<<dim3(INNERD / 64, ROWS / 256), 256, 0, stream>>>(
        qin, Wq, bq, qbuf, ROWS, 2 * DD, INNERD, nullptr, nullptr);

    // 4) projections + softmax + positions + bilinear gather -> y, offsets
    k_sample<<<ROWS, 128, 0, stream>>>(qbuf, vmt, Wr, br, Wd, bd, Ww, bw,
                                       boff, moff, ybuf, out_offs);

    // 5) out = y @ Wo + bo + (x + te1)   (16384 x 256, K=512)
    k_gemm_wmma<<<dim3(DD / 64, ROWS / 256), 256, 0, stream>>>(
        ybuf, Wo, bo, out_main, ROWS, INNERD, DD, x, te + DD);
}